// SubGraph_12627203851073
// MI455X (gfx1250) — compile-verified
//
#include <hip/hip_runtime.h>
#include <hip/hip_bf16.h>

typedef float v2f __attribute__((ext_vector_type(2)));
typedef float v8f __attribute__((ext_vector_type(8)));

#define B_POLY 4096
#define N_VEC 64
#define H_DIM 128
#define H2 64
#define DEPTH 3
#define HS_STRIDE 132   // padded so WMMA fragment b64 LDS loads are bank-conflict-free
#define ENC_STRIDE 66

// LDS layout (in floats)
#define OFF_HS   0
#define OFF_WT   (64*HS_STRIDE)                 // 8448
#define OFF_ENC  (OFF_WT + 64*HS_STRIDE)        // 16896
#define OFF_BIAS (OFF_ENC + 64*ENC_STRIDE)      // 21120
#define OFF_LNW  (OFF_BIAS + 64)
#define OFF_LNB  (OFF_LNW + 64)
#define OFF_M1   (OFF_LNB + 64)
#define OFF_M2   (OFF_M1 + 64)
#define LDS_FLOATS (OFF_M2 + 64)                // 21440 floats = 85760 B

// CDNA5 async global->LDS copy (ASYNCcnt-tracked), 16B per lane.
// vdst = LDS byte offset (per lane), vaddr = 32-bit byte offset, saddr = base.
__device__ __forceinline__ void async_g2l_b128(unsigned ldsOff, unsigned gOff,
                                               const void* base) {
  asm volatile("global_load_async_to_lds_b128 %0, %1, %2 offset:0"
               :: "v"(ldsOff), "v"(gOff), "s"(base) : "memory");
}
__device__ __forceinline__ void wait_asynccnt0() {
  asm volatile("s_wait_asynccnt 0" ::: "memory");
}

__global__ void __launch_bounds__(128)
polyline_encoder_kernel(const float* __restrict__ hidden_states,
                        const int* __restrict__ li_vector_num,
                        const float* __restrict__ Ws,
                        const float* __restrict__ bs,
                        const float* __restrict__ ln_w,
                        const float* __restrict__ ln_b,
                        float* __restrict__ out)
{
  extern __shared__ float smem[];
  float* sHs   = smem + OFF_HS;    // [64][132] current hidden states (128 cols used)
  float* sWt   = smem + OFF_WT;    // [64][132] W transposed: Wt[col][k], 128 k used
  float* sEnc  = smem + OFF_ENC;   // [64][66]  post-GEMM / encoded states
  float* sBias = smem + OFF_BIAS;
  float* sLnW  = smem + OFF_LNW;
  float* sLnB  = smem + OFF_LNB;
  float* sM1   = smem + OFF_M1;
  float* sM2   = smem + OFF_M2;

  const int tid  = threadIdx.x;
  const int b    = blockIdx.x;
  const int lane = tid & 31;
  const int l16  = lane & 15;
  const int hi   = lane >> 4;        // half-wave: K offset +2 for A/B frags, row +8 for C/D
  const int m0   = (tid >> 5) * 16;  // wave's M tile origin
  const int li   = li_vector_num[b];

  // ---- async-copy hidden_states[b] (64x128 f32 = 32KB) straight into LDS ----
  // 2048 x float4; 16 async b128 ops per thread-iteration slot, no VGPR staging.
  {
    const float* gbase = hidden_states + (size_t)b * (N_VEC * H_DIM);
    const unsigned ldsBase = __builtin_amdgcn_groupstaticsize();  // dyn-LDS start
    #pragma unroll 4
    for (int it = 0; it < 16; ++it) {
      int idx4 = it * 128 + tid;
      int r  = idx4 >> 5;       // 32 float4 per row
      int c4 = idx4 & 31;
      unsigned ldsOff = ldsBase + (unsigned)((OFF_HS + r * HS_STRIDE + c4 * 4) * 4);
      async_g2l_b128(ldsOff, (unsigned)(idx4 * 16), gbase);
    }
    wait_asynccnt0();
  }

  for (int l = 0; l < DEPTH; ++l) {
    // ---- stage W[l] transposed into LDS + per-layer params ----
    const float4* w4 = (const float4*)(Ws + (size_t)l * (H_DIM * H2));
    #pragma unroll 4
    for (int it = 0; it < 16; ++it) {
      int idx4 = it * 128 + tid;
      int k  = idx4 >> 4;            // 16 float4 per k-row of 64 cols
      int c0 = (idx4 & 15) * 4;
      float4 w = w4[idx4];
      sWt[(c0 + 0) * HS_STRIDE + k] = w.x;
      sWt[(c0 + 1) * HS_STRIDE + k] = w.y;
      sWt[(c0 + 2) * HS_STRIDE + k] = w.z;
      sWt[(c0 + 3) * HS_STRIDE + k] = w.w;
    }
    if (tid < 64) {
      sBias[tid] = bs[l * H2 + tid];
      sLnW[tid]  = ln_w[l * H2 + tid];
      sLnB[tid]  = ln_b[l * H2 + tid];
    }
    __syncthreads();

    // ---- GEMM: x = hs @ W + bias via V_WMMA_F32_16X16X4_F32 ----
    // wave computes 16x64 strip: 4 N-tiles, K accumulated 4 at a time.
    v8f acc[4];
    #pragma unroll
    for (int nt = 0; nt < 4; ++nt) {
      float bv = sBias[nt * 16 + l16];      // bias depends on column only
      #pragma unroll
      for (int v = 0; v < 8; ++v) acc[nt][v] = bv;
    }
    #pragma unroll 4
    for (int k0 = 0; k0 < H_DIM; k0 += 4) {
      int kb = k0 + 2 * hi;
      // A frag 16x4: lane row = m0+l16, holds K=kb,kb+1 (b64 LDS load)
      v2f a = *(const v2f*)&sHs[(m0 + l16) * HS_STRIDE + kb];
      #pragma unroll
      for (int nt = 0; nt < 4; ++nt) {
        // B frag 4x16: lane col = nt*16+l16, holds K=kb,kb+1 from Wt
        v2f bb = *(const v2f*)&sWt[(nt * 16 + l16) * HS_STRIDE + kb];
        acc[nt] = __builtin_amdgcn_wmma_f32_16x16x4_f32(
            false, a, false, bb, (short)0, acc[nt], false, false);
      }
    }
    // C/D layout: VGPR v -> row m0 + v + 8*hi, col nt*16 + l16
    #pragma unroll
    for (int nt = 0; nt < 4; ++nt) {
      #pragma unroll
      for (int v = 0; v < 8; ++v)
        sEnc[(m0 + v + 8 * hi) * ENC_STRIDE + nt * 16 + l16] = acc[nt][v];
    }
    __syncthreads();

    // ---- LayerNorm + ReLU per row (64 rows, one thread each) ----
    if (tid < 64) {
      float* row = &sEnc[tid * ENC_STRIDE];
      float s = 0.f, ss = 0.f;
      for (int c = 0; c < H2; ++c) { float v = row[c]; s += v; ss += v * v; }
      float mu   = s * (1.f / 64.f);
      float var  = ss * (1.f / 64.f) - mu * mu;
      float rstd = rsqrtf(var + 1e-5f);
      for (int c = 0; c < H2; ++c) {
        float v = sLnW[c] * ((row[c] - mu) * rstd) + sLnB[c];
        row[c] = fmaxf(v, 0.f);
      }
    }
    __syncthreads();

    // ---- masked column top-2 (one thread per column) ----
    if (tid < 64) {
      float m1 = -3.0e38f, m2 = -3.0e38f;
      for (int r = 0; r < N_VEC; ++r) {
        float m = sEnc[r * ENC_STRIDE + tid] + ((r >= li) ? -10000.f : 0.f);
        if (m > m1) { m2 = m1; m1 = m; }
        else if (m > m2) { m2 = m; }
      }
      sM1[tid] = m1; sM2[tid] = m2;
    }
    __syncthreads();

    // ---- hs = concat(enc, leave-one-out max) ----
    #pragma unroll 4
    for (int it = 0; it < 32; ++it) {
      int idx = it * 128 + tid;
      int r = idx >> 6, c = idx & 63;
      float e  = sEnc[r * ENC_STRIDE + c];
      float m  = e + ((r >= li) ? -10000.f : 0.f);
      float M1 = sM1[c], M2 = sM2[c];
      float loo = (m == M1) ? M2 : M1;   // max over i != r (top-2 trick)
      loo = fmaxf(loo, m - 10000.f);     // faithful: self at -10000
      loo = fmaxf(loo, 0.f);             // max(.., 0)
      sHs[r * HS_STRIDE + c]      = e;
      sHs[r * HS_STRIDE + 64 + c] = loo;
    }
    __syncthreads();
  }

  // ---- final max over the 64 rows; thread = output channel ----
  float best = -3.0e38f;
  #pragma unroll 4
  for (int r = 0; r < N_VEC; ++r)
    best = fmaxf(best, sHs[r * HS_STRIDE + tid]);
  out[(size_t)b * H_DIM + tid] = best;
}

extern "C" void kernel_launch(void* const* d_in, const int* in_sizes, int n_in,
                              void* d_out, int out_size, void* d_ws, size_t ws_size,
                              hipStream_t stream) {
  (void)in_sizes; (void)n_in; (void)d_ws; (void)ws_size; (void)out_size;
  const float* hs = (const float*)d_in[0];
  const int*   li = (const int*)d_in[1];
  const float* Ws = (const float*)d_in[2];
  const float* bsp = (const float*)d_in[3];
  const float* lw = (const float*)d_in[4];
  const float* lb = (const float*)d_in[5];
  float* out = (float*)d_out;
  size_t shmem = (size_t)LDS_FLOATS * sizeof(float);
  polyline_encoder_kernel<<<B_POLY, 128, shmem, stream>>>(hs, li, Ws, bsp, lw, lb, out);
}